// SynthPolicyNet_23605140259147
// MI455X (gfx1250) — compile-verified
//
#include <hip/hip_runtime.h>
#include <hip/hip_bf16.h>

typedef __attribute__((ext_vector_type(2))) float v2f;
typedef __attribute__((ext_vector_type(8))) float v8f;

#define HID 256
#define NGR 1024

// ---------------- degree / normalization ----------------

__global__ void deg_kernel(const int* __restrict__ dst, float* __restrict__ deg, int nE) {
  int e = blockIdx.x * blockDim.x + threadIdx.x;
  if (e < nE) atomicAdd(&deg[dst[e]], 1.0f);
}

__global__ void dinv_kernel(float* __restrict__ dinv, int n) {
  int i = blockIdx.x * blockDim.x + threadIdx.x;
  if (i < n) dinv[i] = rsqrtf(dinv[i] + 1.0f);   // +1 = self loop; deg>=1 always
}

// out[i][f] = dinv[i]^2 * H[i][f]   (self-loop message)
__global__ void selfloop_kernel(const float* __restrict__ H, const float* __restrict__ dinv,
                                float* __restrict__ out) {
  size_t idx = (size_t)blockIdx.x * 256u + threadIdx.x;
  float d = dinv[idx >> 8];
  out[idx] = d * d * H[idx];
}

// out[dst] += dinv[src]*dinv[dst] * H[src]  — one block per edge, 256 features
__global__ void scatter_kernel(const float* __restrict__ H, const int* __restrict__ src,
                               const int* __restrict__ dst, const float* __restrict__ dinv,
                               float* __restrict__ out) {
  int e = blockIdx.x;
  int f = threadIdx.x;
  int s = src[e], d = dst[e];
  float w = dinv[s] * dinv[d];
  atomicAdd(&out[(size_t)d * HID + f], w * H[(size_t)s * HID + f]);
}

__global__ void bias_relu_kernel(float* __restrict__ out, const float* __restrict__ b) {
  size_t idx = (size_t)blockIdx.x * 256u + threadIdx.x;
  out[idx] = fmaxf(out[idx] + b[threadIdx.x], 0.0f);
}

// ---------------- pooling ----------------

__global__ void pool_sum_kernel(const float* __restrict__ H, const int* __restrict__ batch,
                                float* __restrict__ sums, float* __restrict__ cnts) {
  int i = blockIdx.x, f = threadIdx.x;
  int g = batch[i];
  atomicAdd(&sums[(size_t)g * HID + f], H[(size_t)i * HID + f]);
  if (f == 0) atomicAdd(&cnts[g], 1.0f);
}

__global__ void pool_div_kernel(const float* __restrict__ sums, const float* __restrict__ cnts,
                                float* __restrict__ hstate) {
  int g = blockIdx.x, f = threadIdx.x;
  hstate[(size_t)g * HID + f] = sums[(size_t)g * HID + f] / fmaxf(cnts[g], 1.0f);
}

// inv[row] = 1 / max(||X[row]||, 1e-12) — one wave per row
__global__ void row_inv_norm_kernel(const float* __restrict__ X, float* __restrict__ inv,
                                    int rows, int cols) {
  int wave = (blockIdx.x * blockDim.x + threadIdx.x) >> 5;
  int lane = threadIdx.x & 31;
  if (wave >= rows) return;
  const float* r = X + (size_t)wave * cols;
  float s = 0.0f;
  for (int j = lane; j < cols; j += 32) { float v = r[j]; s += v * v; }
  #pragma unroll
  for (int m = 16; m; m >>= 1) s += __shfl_xor(s, m, 32);
  if (lane == 0) inv[wave] = 1.0f / fmaxf(sqrtf(s), 1e-12f);
}

__global__ void build_rxn_kernel(const float* __restrict__ hstate, const float* __restrict__ BE,
                                 const int* __restrict__ bidx, float* __restrict__ rxnin) {
  int g = blockIdx.x, k = threadIdx.x; // blockDim = 512
  float v = (k < HID) ? hstate[(size_t)g * HID + k]
                      : BE[(size_t)bidx[g] * HID + (k - HID)];
  rxnin[(size_t)g * (2 * HID) + k] = v;
}

// ---------------- WMMA f32 GEMMs ----------------
// A-frag (16x4): lane = half*16 + m; v0=A[m][k+2*half], v1=A[m][k+2*half+1]
// B-frag (4x16): lane = half*16 + n; v0=B[k+2*half][n], v1=B[k+2*half+1][n]
// C/D (16x16):   acc[r] at lane -> (m = r + 8*half, n = lane&15)

// C(MxN) = A(MxK) @ B(KxN), B row-major; optional bias + relu epilogue.
__global__ void wmma_gemm_kn_kernel(const float* __restrict__ A, const float* __restrict__ B,
                                    float* __restrict__ C, const float* __restrict__ bias,
                                    int M, int N, int K, int applyRelu) {
  const int lane = threadIdx.x & 31;
  const int wave = blockIdx.x * (blockDim.x >> 5) + (threadIdx.x >> 5);
  const int nTiles = N >> 4;
  const int total = (M >> 4) * nTiles;
  if (wave >= total) return;
  const int mT = wave / nTiles, nT = wave % nTiles;
  const int mBase = mT << 4, nBase = nT << 4;
  const int ml = lane & 15, half = lane >> 4;

  const float* aRow = A + (size_t)(mBase + ml) * K + (half << 1);
  const float* bCol = B + nBase + ml;

  v8f acc = {};
  for (int k = 0; k < K; k += 4) {
    v2f a = *(const v2f*)(aRow + k);
    int kk = k + (half << 1);
    v2f b;
    b.x = bCol[(size_t)kk * N];
    b.y = bCol[(size_t)(kk + 1) * N];
    acc = __builtin_amdgcn_wmma_f32_16x16x4_f32(false, a, false, b, (short)0, acc,
                                                false, false);
  }
  float bv = bias ? bias[nBase + ml] : 0.0f;
  #pragma unroll
  for (int r = 0; r < 8; ++r) {
    int m = mBase + r + (half << 3);
    float v = acc[r] + bv;
    if (applyRelu) v = fmaxf(v, 0.0f);
    C[(size_t)m * N + nBase + ml] = v;
  }
}

// C(MxN) = diag(rowInv) * (A(MxK) @ B(NxK)^T) * diag(colInv) / max(temp,1e-4)
// One wave computes a 16x64 strip (4 sub-tiles) to reuse the A fragment.
__global__ void wmma_logits_kernel(const float* __restrict__ A, const float* __restrict__ Bm,
                                   float* __restrict__ C,
                                   const float* __restrict__ rowInv,
                                   const float* __restrict__ colInv,
                                   const float* __restrict__ tptr,
                                   int M, int N, int K) {
  const int lane = threadIdx.x & 31;
  const int wave = blockIdx.x * (blockDim.x >> 5) + (threadIdx.x >> 5);
  const int nTiles = N >> 4;
  const int nGroups = (nTiles + 3) >> 2;
  const int total = (M >> 4) * nGroups;
  if (wave >= total) return;
  const int mT = wave / nGroups, nG = wave % nGroups;
  const int mBase = mT << 4;
  const int ml = lane & 15, half = lane >> 4;
  const float invt = 1.0f / fmaxf(tptr[0], 1e-4f);
  const int nT0 = nG << 2;

  bool valid[4];
  const float* bRow[4];
  #pragma unroll
  for (int s = 0; s < 4; ++s) {
    valid[s] = (nT0 + s) < nTiles;          // wave-uniform
    int row = (nT0 + s) * 16 + ml;
    if (row > N - 1) row = N - 1;           // clamp keeps loads in-bounds
    bRow[s] = Bm + (size_t)row * K + (half << 1);
  }
  const float* aRow = A + (size_t)(mBase + ml) * K + (half << 1);

  v8f acc[4] = {};
  for (int k = 0; k < K; k += 4) {
    v2f a = *(const v2f*)(aRow + k);
    #pragma unroll
    for (int s = 0; s < 4; ++s) {
      v2f b = *(const v2f*)(bRow[s] + k);
      acc[s] = __builtin_amdgcn_wmma_f32_16x16x4_f32(false, a, false, b, (short)0,
                                                     acc[s], false, false);
    }
  }
  #pragma unroll
  for (int s = 0; s < 4; ++s) {
    if (!valid[s]) continue;
    int n = (nT0 + s) * 16 + ml;
    float cs = colInv[n] * invt;
    #pragma unroll
    for (int r = 0; r < 8; ++r) {
      int m = mBase + r + (half << 3);
      C[(size_t)m * N + n] = acc[s][r] * rowInv[m] * cs;
    }
  }
}

// ---------------- host ----------------

extern "C" void kernel_launch(void* const* d_in, const int* in_sizes, int n_in,
                              void* d_out, int out_size, void* d_ws, size_t ws_size,
                              hipStream_t stream) {
  const float* x     = (const float*)d_in[0];
  const int*   ei    = (const int*)d_in[1];
  const int*   batch = (const int*)d_in[2];
  const float* BE    = (const float*)d_in[3];
  const int*   bidx  = (const int*)d_in[4];
  const float* W1  = (const float*)d_in[5];
  const float* b1  = (const float*)d_in[6];
  const float* W2  = (const float*)d_in[7];
  const float* b2  = (const float*)d_in[8];
  const float* W3  = (const float*)d_in[9];
  const float* b3  = (const float*)d_in[10];
  const float* Wr1 = (const float*)d_in[11];
  const float* br1 = (const float*)d_in[12];
  const float* Wr2 = (const float*)d_in[13];
  const float* br2 = (const float*)d_in[14];
  const float* temp = (const float*)d_in[15];

  const int FIN  = 128;
  const int NN   = in_sizes[0] / FIN;   // 200000
  const int E    = in_sizes[1] / 2;     // 800000
  const int NB   = in_sizes[3] / HID;   // 100000
  const int NRXN = in_sizes[14];        // 128
  const int NG   = NGR;                 // 1024

  float* wf = (float*)d_ws;
  size_t o = 0;
  float* bufA   = wf + o; o += (size_t)NN * HID;
  float* bufB   = wf + o; o += (size_t)NN * HID;
  float* dinv   = wf + o; o += (size_t)NN;
  float* sums   = wf + o; o += (size_t)NG * HID;
  float* cnts   = wf + o; o += (size_t)NG;
  float* hstate = wf + o; o += (size_t)NG * HID;
  float* hsinv  = wf + o; o += (size_t)NG;
  float* beinv  = wf + o; o += (size_t)NB;
  float* rxnin  = wf + o; o += (size_t)NG * 2 * HID;
  float* rxnhid = wf + o; o += (size_t)NG * HID;

  auto cdiv = [](long a, long b) { return (int)((a + b - 1) / b); };
  const int* esrc = ei;
  const int* edst = ei + E;

  // degree -> dinv
  hipMemsetAsync(dinv, 0, (size_t)NN * sizeof(float), stream);
  deg_kernel<<<cdiv(E, 256), 256, 0, stream>>>(edst, dinv, E);
  dinv_kernel<<<cdiv(NN, 256), 256, 0, stream>>>(dinv, NN);

  // GCN layer 1
  {
    int waves = (NN / 16) * (HID / 16);
    wmma_gemm_kn_kernel<<<cdiv(waves, 8), 256, 0, stream>>>(x, W1, bufA, nullptr,
                                                            NN, HID, FIN, 0);
    selfloop_kernel<<<NN, 256, 0, stream>>>(bufA, dinv, bufB);
    scatter_kernel<<<E, 256, 0, stream>>>(bufA, esrc, edst, dinv, bufB);
    bias_relu_kernel<<<NN, 256, 0, stream>>>(bufB, b1);
  }
  // GCN layers 2 & 3
  const float* Ws[2] = {W2, W3};
  const float* bs[2] = {b2, b3};
  for (int l = 0; l < 2; ++l) {
    int waves = (NN / 16) * (HID / 16);
    wmma_gemm_kn_kernel<<<cdiv(waves, 8), 256, 0, stream>>>(bufB, Ws[l], bufA, nullptr,
                                                            NN, HID, HID, 0);
    selfloop_kernel<<<NN, 256, 0, stream>>>(bufA, dinv, bufB);
    scatter_kernel<<<E, 256, 0, stream>>>(bufA, esrc, edst, dinv, bufB);
    bias_relu_kernel<<<NN, 256, 0, stream>>>(bufB, bs[l]);
  }

  // global mean pool
  hipMemsetAsync(sums, 0, (size_t)NG * HID * sizeof(float), stream);
  hipMemsetAsync(cnts, 0, (size_t)NG * sizeof(float), stream);
  pool_sum_kernel<<<NN, 256, 0, stream>>>(bufB, batch, sums, cnts);
  pool_div_kernel<<<NG, 256, 0, stream>>>(sums, cnts, hstate);

  // inverse norms for cosine similarity
  row_inv_norm_kernel<<<cdiv(NG, 8), 256, 0, stream>>>(hstate, hsinv, NG, HID);
  row_inv_norm_kernel<<<cdiv(NB, 8), 256, 0, stream>>>(BE, beinv, NB, HID);

  float* outf = (float*)d_out;
  // block logits: (1024 x 100000)
  {
    int nTiles = NB / 16;
    int nGroups = (nTiles + 3) / 4;
    int waves = (NG / 16) * nGroups;
    wmma_logits_kernel<<<cdiv(waves, 8), 256, 0, stream>>>(hstate, BE, outf,
                                                           hsinv, beinv, temp,
                                                           NG, NB, HID);
  }
  // reaction head
  build_rxn_kernel<<<NG, 512, 0, stream>>>(hstate, BE, bidx, rxnin);
  {
    int waves = (NG / 16) * (HID / 16);
    wmma_gemm_kn_kernel<<<cdiv(waves, 8), 256, 0, stream>>>(rxnin, Wr1, rxnhid, br1,
                                                            NG, HID, 2 * HID, 1);
  }
  {
    int waves = (NG / 16) * (NRXN / 16);
    wmma_gemm_kn_kernel<<<cdiv(waves, 8), 256, 0, stream>>>(rxnhid, Wr2,
                                                            outf + (size_t)NG * NB, br2,
                                                            NG, NRXN, HID, 0);
  }
}